// KcompetitiveLayer_60043642798719
// MI455X (gfx1250) — compile-verified
//
#include <hip/hip_runtime.h>
#include <stdint.h>

// ---------------- constants ----------------
#define KP      64        // ceil(K/2) positive winners
#define KN      64        // floor(K/2)
#define AA      6.26f
#define NBIN    1024      // histogram over top 11 bits of a positive float (sign=0)
#define HCOPIES 4         // replicated sub-histograms to cut LDS atomic conflicts
#define HSTRIDE (NBIN + 8) // pad: copy stride 1032 words -> +8 bank shift per copy
#define CAP     8192      // candidate buffer capacity per side
#define RBLK    2048      // fixed grid for streaming kernels

typedef float v4f __attribute__((ext_vector_type(4)));

// ---------------- workspace layout ----------------
struct WS {
  unsigned hist_pos[NBIN];
  unsigned hist_neg[NBIN];
  float    blk_sp[RBLK];
  float    blk_sn[RBLK];
  unsigned blk_cp[RBLK];
  unsigned blk_cn[RBLK];
  unsigned n_pos, n_neg;
  float    sum_pos, sum_absneg;
  unsigned pos_thresh, neg_thresh;
  unsigned pos_cnt, neg_cnt;
  unsigned cond_pos, cond_neg;
  float    e_pos, e_neg;
  unsigned kth_idx;
  unsigned winners[KP];
  unsigned long long pos_cand[CAP];
  unsigned long long neg_cand[CAP];
};

// ---------------- CDNA5 async global->LDS helpers (ASYNCcnt-tracked DMA) ------
__device__ __forceinline__ void async_load_b128(unsigned lds_off, const void* gptr) {
  unsigned long long ga = (unsigned long long)(uintptr_t)gptr;
  // GLOBAL_LOAD_ASYNC_TO_LDS_B128 (GV mode): VDST = LDS byte-offset VGPR,
  // VADDR = 64-bit global address pair.
  asm volatile("global_load_async_to_lds_b128 %0, %1, off"
               :: "v"(lds_off), "v"(ga)
               : "memory");
}
__device__ __forceinline__ void wait_async_le1() {
  asm volatile("s_wait_asynccnt 0x1" ::: "memory");
}
__device__ __forceinline__ void wait_async_le0() {
  asm volatile("s_wait_asynccnt 0x0" ::: "memory");
}

// ---------------- kernel 0: zero counters/histograms ----------------
__global__ void zero_kernel(WS* ws) {
  int gid = blockIdx.x * blockDim.x + threadIdx.x;
  int gstride = gridDim.x * blockDim.x;
  for (int b = gid; b < NBIN; b += gstride) {
    ws->hist_pos[b] = 0u;
    ws->hist_neg[b] = 0u;
  }
  if (gid == 0) { ws->pos_cnt = 0u; ws->neg_cnt = 0u; }
}

// ---------------- kernel 1: histogram + per-block partial stats ----------------
// Streams x once through the async global->LDS double-buffered pipeline.
// Histogram uses 4 bank-padded replicas to cut same-bin atomic serialization.
__global__ __launch_bounds__(256) void hist_kernel(const float* __restrict__ x,
                                                   long long n, WS* __restrict__ ws) {
  __shared__ v4f      sbuf[2][256];
  __shared__ unsigned lhp[HCOPIES * HSTRIDE];
  __shared__ unsigned lhn[HCOPIES * HSTRIDE];
  __shared__ float    fred[256];
  __shared__ unsigned ured[256];
  const int tid = threadIdx.x;
  for (int b = tid; b < HCOPIES * HSTRIDE; b += 256) { lhp[b] = 0u; lhn[b] = 0u; }
  __syncthreads();

  const unsigned hoff = (unsigned)(tid & (HCOPIES - 1)) * HSTRIDE;
  unsigned cp = 0u, cn = 0u;
  float sp = 0.f, sn = 0.f;

  const v4f* x4 = (const v4f*)x;
  const long long nt4 = n >> 2;
  const long long ntiles = nt4 >> 8;          // tiles of 256 float4
  const long long stride = gridDim.x;

  unsigned lds0 = (unsigned)(uintptr_t)&sbuf[0][tid];
  unsigned lds1 = (unsigned)(uintptr_t)&sbuf[1][tid];

  long long t = blockIdx.x;
  int par = 0;
  if (t < ntiles) async_load_b128(lds0, &x4[t * 256 + tid]);

  for (; t < ntiles; t += stride) {
    long long tnx = t + stride;
    if (tnx < ntiles) {
      async_load_b128(par ? lds0 : lds1, &x4[tnx * 256 + tid]);  // prefetch next tile
      wait_async_le1();                                          // current tile landed
    } else {
      wait_async_le0();
    }
    v4f v = sbuf[par][tid];
#pragma unroll
    for (int c = 0; c < 4; ++c) {
      float f = v[c];
      unsigned bits = __float_as_uint(f);
      if (f > 0.f)      { cp++; sp += f; atomicAdd(&lhp[hoff + (bits >> 21)], 1u); }
      else if (f < 0.f) { unsigned ab = bits & 0x7fffffffu;
                          cn++; sn += __uint_as_float(ab);
                          atomicAdd(&lhn[hoff + (ab >> 21)], 1u); }
    }
    par ^= 1;
  }
  // scalar tail (elements beyond full tiles)
  const long long done = ntiles << 10;  // ntiles * 256 * 4
  const long long gid = (long long)blockIdx.x * 256 + tid;
  const long long gstride = (long long)gridDim.x * 256;
  for (long long i = done + gid; i < n; i += gstride) {
    float f = x[i];
    unsigned bits = __float_as_uint(f);
    if (f > 0.f)      { cp++; sp += f; atomicAdd(&lhp[hoff + (bits >> 21)], 1u); }
    else if (f < 0.f) { unsigned ab = bits & 0x7fffffffu;
                        cn++; sn += __uint_as_float(ab);
                        atomicAdd(&lhn[hoff + (ab >> 21)], 1u); }
  }
  __syncthreads();
  // fold replicas, flush to global histogram
  for (int b = tid; b < NBIN; b += 256) {
    unsigned p = 0u, q = 0u;
#pragma unroll
    for (int c = 0; c < HCOPIES; ++c) { p += lhp[c * HSTRIDE + b]; q += lhn[c * HSTRIDE + b]; }
    if (p) atomicAdd(&ws->hist_pos[b], p);
    if (q) atomicAdd(&ws->hist_neg[b], q);
  }

  // deterministic block reduction of (sp,cp), then (sn,cn)
  fred[tid] = sp; ured[tid] = cp; __syncthreads();
  for (int s = 128; s > 0; s >>= 1) {
    if (tid < s) { fred[tid] += fred[tid + s]; ured[tid] += ured[tid + s]; }
    __syncthreads();
  }
  if (tid == 0) { ws->blk_sp[blockIdx.x] = fred[0]; ws->blk_cp[blockIdx.x] = ured[0]; }
  __syncthreads();
  fred[tid] = sn; ured[tid] = cn; __syncthreads();
  for (int s = 128; s > 0; s >>= 1) {
    if (tid < s) { fred[tid] += fred[tid + s]; ured[tid] += ured[tid + s]; }
    __syncthreads();
  }
  if (tid == 0) { ws->blk_sn[blockIdx.x] = fred[0]; ws->blk_cn[blockIdx.x] = ured[0]; }
}

// ---------------- kernel 2: final reduction + threshold scan ----------------
__global__ __launch_bounds__(1024) void thresh_kernel(WS* __restrict__ ws) {
  __shared__ float    fr[1024];
  __shared__ unsigned ur[1024];
  const int tid = threadIdx.x;

  fr[tid] = ws->blk_sp[tid] + ws->blk_sp[tid + 1024];
  ur[tid] = ws->blk_cp[tid] + ws->blk_cp[tid + 1024];
  __syncthreads();
  for (int s = 512; s > 0; s >>= 1) {
    if (tid < s) { fr[tid] += fr[tid + s]; ur[tid] += ur[tid + s]; }
    __syncthreads();
  }
  if (tid == 0) { ws->sum_pos = fr[0]; ws->n_pos = ur[0]; ws->cond_pos = (ur[0] > KP) ? 1u : 0u; }
  __syncthreads();

  fr[tid] = ws->blk_sn[tid] + ws->blk_sn[tid + 1024];
  ur[tid] = ws->blk_cn[tid] + ws->blk_cn[tid + 1024];
  __syncthreads();
  for (int s = 512; s > 0; s >>= 1) {
    if (tid < s) { fr[tid] += fr[tid + s]; ur[tid] += ur[tid + s]; }
    __syncthreads();
  }
  if (tid == 0) { ws->sum_absneg = fr[0]; ws->n_neg = ur[0]; ws->cond_neg = (ur[0] > KN) ? 1u : 0u; }

  if (tid == 0) {
    unsigned acc = 0u; unsigned t = 0u;
    for (int b = NBIN - 1; b >= 0; --b) {
      acc += ws->hist_pos[b];
      if (acc >= KP) { t = (unsigned)b; break; }
    }
    ws->pos_thresh = t;
    acc = 0u; t = 0u;
    for (int b = NBIN - 1; b >= 0; --b) {
      acc += ws->hist_neg[b];
      if (acc >= KN) { t = (unsigned)b; break; }
    }
    ws->neg_thresh = t;
  }
}

// ---------------- kernel 3: FUSED candidate collect + elementwise zeroing ------
// Single streaming pass: async-LDS staged read of x, NT store of the zeroed
// output, atomic push of top candidates. Winners are patched by scatter later.
__global__ __launch_bounds__(256) void sweep_kernel(const float* __restrict__ x,
                                                    float* __restrict__ out,
                                                    long long n, WS* __restrict__ ws) {
  __shared__ v4f sbuf[2][256];
  const int tid = threadIdx.x;
  const unsigned tp = ws->pos_thresh;
  const unsigned tn = ws->neg_thresh;
  const bool condp = ws->cond_pos != 0u;
  const bool condn = ws->cond_neg != 0u;

  const v4f* x4 = (const v4f*)x;
  v4f* o4 = (v4f*)out;
  const long long nt4 = n >> 2;
  const long long ntiles = nt4 >> 8;
  const long long stride = gridDim.x;

  unsigned lds0 = (unsigned)(uintptr_t)&sbuf[0][tid];
  unsigned lds1 = (unsigned)(uintptr_t)&sbuf[1][tid];

  long long t = blockIdx.x;
  int par = 0;
  if (t < ntiles) async_load_b128(lds0, &x4[t * 256 + tid]);

  for (; t < ntiles; t += stride) {
    long long tnx = t + stride;
    if (tnx < ntiles) {
      async_load_b128(par ? lds0 : lds1, &x4[tnx * 256 + tid]);
      wait_async_le1();
    } else {
      wait_async_le0();
    }
    v4f v = sbuf[par][tid];
    const long long i4 = t * 256 + tid;
    v4f r;
#pragma unroll
    for (int c = 0; c < 4; ++c) {
      float f = v[c];
      unsigned bits = __float_as_uint(f);
      unsigned idx = (unsigned)(i4 * 4 + c);
      float o = f;
      if (f > 0.f) {
        if (condp) o = 0.f;
        if ((bits >> 21) >= tp) {
          unsigned slot = atomicAdd(&ws->pos_cnt, 1u);
          if (slot < CAP) ws->pos_cand[slot] = ((unsigned long long)bits << 32) | idx;
        }
      } else if (f < 0.f) {
        if (condn) o = 0.f;
        unsigned ab = bits & 0x7fffffffu;
        if ((ab >> 21) >= tn) {
          unsigned slot = atomicAdd(&ws->neg_cnt, 1u);
          if (slot < CAP) ws->neg_cand[slot] = ((unsigned long long)ab << 32) | idx;
        }
      }
      r[c] = o;
    }
    __builtin_nontemporal_store(r, &o4[i4]);
    par ^= 1;
  }

  // scalar tail
  const long long done = ntiles << 10;
  const long long gid = (long long)blockIdx.x * 256 + tid;
  const long long gstride = (long long)gridDim.x * 256;
  for (long long i = done + gid; i < n; i += gstride) {
    float f = x[i];
    unsigned bits = __float_as_uint(f);
    float o = f;
    if (f > 0.f) {
      if (condp) o = 0.f;
      if ((bits >> 21) >= tp) {
        unsigned slot = atomicAdd(&ws->pos_cnt, 1u);
        if (slot < CAP) ws->pos_cand[slot] = ((unsigned long long)bits << 32) | (unsigned)i;
      }
    } else if (f < 0.f) {
      if (condn) o = 0.f;
      unsigned ab = bits & 0x7fffffffu;
      if ((ab >> 21) >= tn) {
        unsigned slot = atomicAdd(&ws->neg_cnt, 1u);
        if (slot < CAP) ws->neg_cand[slot] = ((unsigned long long)ab << 32) | (unsigned)i;
      }
    }
    out[i] = o;
  }
}

// ---------------- kernel 4: exact top-64 selection (single block) ----------------
__global__ __launch_bounds__(1024) void select_kernel(WS* __restrict__ ws) {
  __shared__ unsigned long long red[1024];
  const int tid = threadIdx.x;

  // ---- positive side ----
  {
    unsigned nc = ws->pos_cnt; if (nc > CAP) nc = CAP;
    if (ws->cond_pos) {
      float sumtop = 0.f;
      for (int k = 0; k < KP; ++k) {
        unsigned long long m = 0ull;
        for (unsigned j = tid; j < nc; j += 1024) {
          unsigned long long c = ws->pos_cand[j];
          if (c > m) m = c;
        }
        red[tid] = m; __syncthreads();
        for (int s = 512; s > 0; s >>= 1) {
          if (tid < s && red[tid + s] > red[tid]) red[tid] = red[tid + s];
          __syncthreads();
        }
        unsigned long long w = red[0];
        if (tid == 0) {
          ws->winners[k] = (unsigned)(w & 0xffffffffu);
          sumtop += __uint_as_float((unsigned)(w >> 32));
        }
        for (unsigned j = tid; j < nc; j += 1024)
          if (ws->pos_cand[j] == w) ws->pos_cand[j] = 0ull;
        __syncthreads();
      }
      if (tid == 0) ws->e_pos = AA * (ws->sum_pos - sumtop);
    } else if (tid == 0) {
      ws->e_pos = 0.f;
    }
  }
  __syncthreads();

  // ---- negative side ----
  {
    unsigned nc = ws->neg_cnt; if (nc > CAP) nc = CAP;
    if (ws->cond_neg) {
      float sumtop = 0.f;
      for (int k = 0; k < KN; ++k) {
        unsigned long long m = 0ull;
        for (unsigned j = tid; j < nc; j += 1024) {
          unsigned long long c = ws->neg_cand[j];
          if (c > m) m = c;
        }
        red[tid] = m; __syncthreads();
        for (int s = 512; s > 0; s >>= 1) {
          if (tid < s && red[tid + s] > red[tid]) red[tid] = red[tid + s];
          __syncthreads();
        }
        unsigned long long w = red[0];
        if (tid == 0) {
          sumtop += __uint_as_float((unsigned)(w >> 32));
          if (k == KN - 1) ws->kth_idx = (unsigned)(w & 0xffffffffu);
        }
        for (unsigned j = tid; j < nc; j += 1024)
          if (ws->neg_cand[j] == w) ws->neg_cand[j] = 0ull;
        __syncthreads();
      }
      if (tid == 0) ws->e_neg = -AA * (ws->sum_absneg - sumtop);
    } else if (tid == 0) {
      ws->e_neg = 0.f;
    }
  }
}

// ---------------- kernel 5: scatter the 65 surviving elements ----------------
__global__ void scatter_kernel(const float* __restrict__ x, float* __restrict__ out,
                               const WS* __restrict__ ws) {
  const int t = threadIdx.x;
  if (ws->cond_pos && t < KP) {
    unsigned i = ws->winners[t];
    out[i] = x[i] + ws->e_pos;
  }
  if (ws->cond_neg && t == KP) {
    unsigned i = ws->kth_idx;
    out[i] = x[i] + ws->e_neg;
  }
}

// ---------------- host entry ----------------
extern "C" void kernel_launch(void* const* d_in, const int* in_sizes, int n_in,
                              void* d_out, int out_size, void* d_ws, size_t ws_size,
                              hipStream_t stream) {
  (void)n_in; (void)out_size; (void)ws_size;
  const float* x = (const float*)d_in[0];
  float* out = (float*)d_out;
  WS* ws = (WS*)d_ws;
  const long long n = (long long)in_sizes[0];

  zero_kernel<<<4, 256, 0, stream>>>(ws);
  hist_kernel<<<RBLK, 256, 0, stream>>>(x, n, ws);
  thresh_kernel<<<1, 1024, 0, stream>>>(ws);
  sweep_kernel<<<RBLK, 256, 0, stream>>>(x, out, n, ws);
  select_kernel<<<1, 1024, 0, stream>>>(ws);
  scatter_kernel<<<1, 128, 0, stream>>>(x, out, ws);
}